// FeCLoss_7121055776861
// MI455X (gfx1250) — compile-verified
//
#include <hip/hip_runtime.h>
#include <math.h>

// ---------------------------------------------------------------------------
// FeCLoss fused pipeline for gfx1250 (wave32, WMMA f32 16x16x4)
//   Fw  = normalize(upsample2x(feat)) in WMMA-fragment layout (lane's 16
//         floats contiguous -> one v16f load = 4 clustered b128 loads).
//   GEMM kernels: ONE WAVE PER WORKGROUP so all tile control flow is scalar
//   (s_cbranch, no exec masking), 2x-unrolled ping-pong load buffers (no
//   v_mov rotation), diagonal handled only on its two tiles.
// ---------------------------------------------------------------------------

typedef float v2f  __attribute__((ext_vector_type(2)));
typedef float v8f  __attribute__((ext_vector_type(8)));
typedef float v16f __attribute__((ext_vector_type(16)));

#define NPIX 4096
#define CH   32
#define NT   256            // NPIX / 16 column tiles
#define NRT  128            // NPIX / 32 row strips (32 rows per wave)
#define INV_T 1.6666666666666667f

// ---- helpers --------------------------------------------------------------

__device__ __forceinline__ float rsum16(float x) {
#pragma unroll
    for (int m = 1; m < 16; m <<= 1) x += __shfl_xor(x, m, 32);
    return x;
}

__device__ __forceinline__ float rmax16(float x) {
#pragma unroll
    for (int m = 1; m < 16; m <<= 1) x = fmaxf(x, __shfl_xor(x, m, 32));
    return x;
}

__device__ __forceinline__ v16f load_frag(const float* __restrict__ Fwb,
                                          int tile, int lane) {
    return *(const v16f*)(Fwb + ((size_t)tile * 32 + lane) * 16);
}

// 8 chained WMMAs: full K=32 dot for a 16x16 tile
__device__ __forceinline__ v8f chain8(const v16f& A, const v16f& Bf) {
    v8f c = {0.f, 0.f, 0.f, 0.f, 0.f, 0.f, 0.f, 0.f};
#pragma unroll
    for (int kc = 0; kc < 8; ++kc) {
        v2f aa = { A[2 * kc], A[2 * kc + 1] };
        v2f bb = { Bf[2 * kc], Bf[2 * kc + 1] };
        c = __builtin_amdgcn_wmma_f32_16x16x4_f32(false, aa, false, bb,
                                                  (short)0, c, false, false);
    }
    return c;
}

// ---- kernel 1: upsample + normalize + mask pool -> fragment layout --------

__global__ void prep_kernel(const float* __restrict__ feat,
                            const float* __restrict__ mask,
                            float* __restrict__ Fw,
                            int* __restrict__ cnt) {
    int idx = blockIdx.x * blockDim.x + threadIdx.x;   // 0 .. 8191
    int b = idx >> 12;
    int n = idx & (NPIX - 1);
    int oy = n >> 6, ox = n & 63;

    float yf = (float)oy * (31.0f / 63.0f);
    float xf = (float)ox * (31.0f / 63.0f);
    int y0 = (int)floorf(yf);  int y1 = min(y0 + 1, 31);  float wy = yf - (float)y0;
    int x0 = (int)floorf(xf);  int x1 = min(x0 + 1, 31);  float wx = xf - (float)x0;

    const float* fb = feat + (size_t)b * CH * 1024;
    float vals[CH];
    float ss = 0.f;
#pragma unroll
    for (int c = 0; c < CH; ++c) {
        const float* p = fb + (size_t)c * 1024;
        float v00 = p[y0 * 32 + x0], v01 = p[y0 * 32 + x1];
        float v10 = p[y1 * 32 + x0], v11 = p[y1 * 32 + x1];
        float r0 = v00 + (v10 - v00) * wy;     // y-interp first (matches ref)
        float r1 = v01 + (v11 - v01) * wy;
        float v  = r0 + (r1 - r0) * wx;
        vals[c] = v;
        ss += v * v;
    }
    float inv = 1.0f / fmaxf(sqrtf(ss), 1e-12f);

    float* Fwb = Fw + (size_t)b * NPIX * CH;
    int tile = n >> 4, li = n & 15;
    float* lo = Fwb + ((size_t)(tile * 32 + li)) * 16;
    float* hi = Fwb + ((size_t)(tile * 32 + li + 16)) * 16;
#pragma unroll
    for (int kc = 0; kc < 8; ++kc) {
        v2f lv = { vals[4 * kc + 0] * inv, vals[4 * kc + 1] * inv };
        v2f hv = { vals[4 * kc + 2] * inv, vals[4 * kc + 3] * inv };
        *(v2f*)(lo + 2 * kc) = lv;
        *(v2f*)(hi + 2 * kc) = hv;
    }

    const float* mb = mask + (size_t)b * 512 * 512;
    float s = 0.f;
#pragma unroll
    for (int i = 0; i < 8; ++i) {
        const float* r = mb + (size_t)(oy * 8 + i) * 512 + ox * 8;
#pragma unroll
        for (int j = 0; j < 8; ++j) s += r[j];
    }
    cnt[idx] = (int)(s + 0.5f);
}

// ---- kernel 2: per-row max of masked logits (one wave per workgroup) ------

__global__ void __launch_bounds__(32)
lmax_kernel(const float* __restrict__ Fw, float* __restrict__ lmax) {
    int b = blockIdx.y;
    int strip = blockIdx.x;                  // scalar: uniform control flow
    int lane = threadIdx.x;
    int lm16 = lane & 15, hi = lane >> 4;
    int rowBase = strip * 32;
    int dt0 = strip * 2, dt1 = dt0 + 1;      // the two diagonal column tiles

    const float* Fwb = Fw + (size_t)b * NPIX * CH;
    v16f A0 = load_frag(Fwb, dt0, lane);
    v16f A1 = load_frag(Fwb, dt1, lane);

    float mx[16];
#pragma unroll
    for (int v = 0; v < 16; ++v) mx[v] = 0.0f;   // zeroed diagonal participates

    auto proc = [&](int ct, const v16f& Bf) {
        v8f c0 = chain8(A0, Bf);
        v8f c1 = chain8(A1, Bf);
        if (ct == dt0) {                 // scalar branch: diag in rows 0..15
#pragma unroll
            for (int v = 0; v < 8; ++v) {
                if ((v + hi * 8) != lm16) mx[v] = fmaxf(mx[v], c0[v]);
                mx[v + 8] = fmaxf(mx[v + 8], c1[v]);
            }
        } else if (ct == dt1) {          // scalar branch: diag in rows 16..31
#pragma unroll
            for (int v = 0; v < 8; ++v) {
                mx[v] = fmaxf(mx[v], c0[v]);
                if ((v + hi * 8) != lm16) mx[v + 8] = fmaxf(mx[v + 8], c1[v]);
            }
        } else {                         // compare-free fast path (254 tiles)
#pragma unroll
            for (int v = 0; v < 8; ++v) {
                mx[v]     = fmaxf(mx[v],     c0[v]);
                mx[v + 8] = fmaxf(mx[v + 8], c1[v]);
            }
        }
    };

    v16f Ba = load_frag(Fwb, 0, lane);
    for (int ct = 0; ct < NT; ct += 2) {     // 2x unroll: no buffer copies
        v16f Bb = load_frag(Fwb, ct + 1, lane);
        proc(ct, Ba);
        Ba = load_frag(Fwb, (ct + 2) & (NT - 1), lane);
        proc(ct + 1, Bb);
    }

#pragma unroll
    for (int v = 0; v < 16; ++v) mx[v] = rmax16(mx[v]);
    if (lm16 == 0) {
#pragma unroll
        for (int v = 0; v < 8; ++v) {
            lmax[(size_t)b * NPIX + rowBase + v + hi * 8]      = mx[v] * INV_T;
            lmax[(size_t)b * NPIX + rowBase + 16 + v + hi * 8] = mx[v + 8] * INV_T;
        }
    }
}

// ---- kernel 3: two-pass (neg-sum, then loss), one wave per workgroup ------

__global__ void __launch_bounds__(32)
loss_kernel(const float* __restrict__ Fw,
            const int* __restrict__ cnt,
            const float* __restrict__ lmax,
            float* __restrict__ partials) {
    int b = blockIdx.y;
    int strip = blockIdx.x;                  // scalar
    int lane = threadIdx.x;
    int lm16 = lane & 15, hi = lane >> 4;
    int rowBase = strip * 32;
    int dt0 = strip * 2, dt1 = dt0 + 1;

    const float* Fwb = Fw + (size_t)b * NPIX * CH;
    const int*   cb  = cnt + (size_t)b * NPIX;
    const float* lb  = lmax + (size_t)b * NPIX;

    v16f A0 = load_frag(Fwb, dt0, lane);
    v16f A1 = load_frag(Fwb, dt1, lane);

    int cr[16];
#pragma unroll
    for (int v = 0; v < 8; ++v) {
        cr[v]     = cb[rowBase + v + hi * 8];
        cr[v + 8] = cb[rowBase + 16 + v + hi * 8];
    }

    float ns[16], pos[16];
#pragma unroll
    for (int v = 0; v < 16; ++v) { ns[v] = 0.f; pos[v] = 0.f; }

    // ---- pass 1: neg_sum and positive count per row (diag is a positive) ----
    {
        auto proc = [&](const v16f& Bf, int cc, float lmc) {
            v8f c0 = chain8(A0, Bf);
            v8f c1 = chain8(A1, Bf);
#pragma unroll
            for (int v = 0; v < 8; ++v) {
                float e0 = expf(c0[v] * INV_T - lmc);
                float e1 = expf(c1[v] * INV_T - lmc);
                if (cr[v] == cc)     pos[v]     += 1.0f;  else ns[v]     += e0;
                if (cr[v + 8] == cc) pos[v + 8] += 1.0f;  else ns[v + 8] += e1;
            }
        };
        v16f  Ba  = load_frag(Fwb, 0, lane);
        int   ca  = cb[lm16];
        float la  = lb[lm16];
        for (int ct = 0; ct < NT; ct += 2) {
            v16f  Bb = load_frag(Fwb, ct + 1, lane);
            int   cbv = cb[(ct + 1) * 16 + lm16];
            float lbv = lb[(ct + 1) * 16 + lm16];
            proc(Ba, ca, la);
            int ctn = (ct + 2) & (NT - 1);
            Ba = load_frag(Fwb, ctn, lane);
            ca = cb[ctn * 16 + lm16];
            la = lb[ctn * 16 + lm16];
            proc(Bb, cbv, lbv);
        }
    }
#pragma unroll
    for (int v = 0; v < 16; ++v) { ns[v] = rsum16(ns[v]); pos[v] = rsum16(pos[v]); }

    // ---- pass 2: loss terms ----
    float loss[16];
#pragma unroll
    for (int v = 0; v < 16; ++v) loss[v] = 0.f;

    {
        auto proc = [&](int ct, const v16f& Bf, int cc, float lmc) {
            v8f c0 = chain8(A0, Bf);
            v8f c1 = chain8(A1, Bf);
            if (ct == dt0) {             // scalar branch
#pragma unroll
                for (int v = 0; v < 8; ++v) {
                    float sh0 = c0[v] * INV_T - lmc;
                    float sh1 = c1[v] * INV_T - lmc;
                    bool diag = (v + hi * 8) == lm16;
                    if (!diag && cr[v] == cc)
                        loss[v]     += logf(expf(sh0) + ns[v])     - sh0;
                    if (cr[v + 8] == cc)
                        loss[v + 8] += logf(expf(sh1) + ns[v + 8]) - sh1;
                }
            } else if (ct == dt1) {      // scalar branch
#pragma unroll
                for (int v = 0; v < 8; ++v) {
                    float sh0 = c0[v] * INV_T - lmc;
                    float sh1 = c1[v] * INV_T - lmc;
                    bool diag = (v + hi * 8) == lm16;
                    if (cr[v] == cc)
                        loss[v]     += logf(expf(sh0) + ns[v])     - sh0;
                    if (!diag && cr[v + 8] == cc)
                        loss[v + 8] += logf(expf(sh1) + ns[v + 8]) - sh1;
                }
            } else {                     // no-diag fast path (254 tiles)
#pragma unroll
                for (int v = 0; v < 8; ++v) {
                    float sh0 = c0[v] * INV_T - lmc;
                    float sh1 = c1[v] * INV_T - lmc;
                    if (cr[v] == cc)
                        loss[v]     += logf(expf(sh0) + ns[v])     - sh0;
                    if (cr[v + 8] == cc)
                        loss[v + 8] += logf(expf(sh1) + ns[v + 8]) - sh1;
                }
            }
        };
        v16f  Ba  = load_frag(Fwb, 0, lane);
        int   ca  = cb[lm16];
        float la  = lb[lm16];
        for (int ct = 0; ct < NT; ct += 2) {
            v16f  Bb = load_frag(Fwb, ct + 1, lane);
            int   cbv = cb[(ct + 1) * 16 + lm16];
            float lbv = lb[(ct + 1) * 16 + lm16];
            proc(ct, Ba, ca, la);
            int ctn = (ct + 2) & (NT - 1);
            Ba = load_frag(Fwb, ctn, lane);
            ca = cb[ctn * 16 + lm16];
            la = lb[ctn * 16 + lm16];
            proc(ct + 1, Bb, cbv, lbv);
        }
    }
#pragma unroll
    for (int v = 0; v < 16; ++v) loss[v] = rsum16(loss[v]);

    float stripSum = 0.f;
    if (lm16 == 0) {
#pragma unroll
        for (int v = 0; v < 16; ++v)
            stripSum += loss[v] / (pos[v] - 1.0f + 1e-18f);
    }
    stripSum += __shfl_xor(stripSum, 16, 32);   // combine the two lane halves
    if (lane == 0) partials[(size_t)b * NRT + strip] = stripSum;
}

// ---- kernel 4: deterministic tree reduction of 256 strip partials ---------

__global__ void reduce_kernel(const float* __restrict__ partials,
                              float* __restrict__ out) {
    __shared__ float s[256];
    int t = threadIdx.x;
    s[t] = partials[t];
    __syncthreads();
#pragma unroll
    for (int st = 128; st > 0; st >>= 1) {
        if (t < st) s[t] += s[t + st];
        __syncthreads();
    }
    if (t == 0) out[0] = s[0] / (2.0f * (float)NPIX);
}

// ---------------------------------------------------------------------------

extern "C" void kernel_launch(void* const* d_in, const int* in_sizes, int n_in,
                              void* d_out, int out_size, void* d_ws, size_t ws_size,
                              hipStream_t stream) {
    const float* feat = (const float*)d_in[0];   // (2,32,32,32)
    const float* mask = (const float*)d_in[1];   // (2,1,512,512)
    float* out = (float*)d_out;                  // scalar loss

    float* Fw       = (float*)d_ws;              // 2*4096*32 floats (frag layout)
    int*   cnt      = (int*)(Fw + 2 * NPIX * CH);// 8192 ints
    float* lmax     = (float*)(cnt + 2 * NPIX);  // 8192 floats
    float* partials = lmax + 2 * NPIX;           // 256 floats

    prep_kernel<<<dim3(32), dim3(256), 0, stream>>>(feat, mask, Fw, cnt);

    dim3 grid(NRT, 2);          // one wave per workgroup: uniform control flow
    lmax_kernel<<<grid, dim3(32), 0, stream>>>(Fw, lmax);
    loss_kernel<<<grid, dim3(32), 0, stream>>>(Fw, cnt, lmax, partials);

    reduce_kernel<<<dim3(1), dim3(256), 0, stream>>>(partials, out);
}